// ListMLE_loss_tail_48232482734819
// MI455X (gfx1250) — compile-verified
//
#include <hip/hip_runtime.h>
#include <hip/hip_bf16.h>
#include <stdint.h>

// ---------------------------------------------------------------------------
// ListMLE-with-tail loss, MI455X (gfx1250).
// HBM-bound streaming reduction: 1024*50000*4B = 204.8 MB read
//   -> ~8.8 us floor @ 23.3 TB/s.
// Data path: Tensor Data Mover (tensor_load_to_lds), 4-deep LDS double
// buffering (3 chunks = 48 KB in flight), s_wait_tensorcnt for exact
// in-order completion, one s_barrier per chunk. VALU does v_exp_f32 + adds.
// Tail term fully parallelized across lanes (no serial load chains).
// ---------------------------------------------------------------------------

#define CHUNK 4096   // floats per TDM tile (16 KB)
#define NBUF  4      // 64 KB LDS ring
#define TMAX  64     // >= T (=50)

typedef __attribute__((ext_vector_type(4))) unsigned int u32x4;
typedef __attribute__((ext_vector_type(8))) unsigned int u32x8;

// Issue a 1-D TDM load of `nelem` f32 from `gptr` to LDS byte offset
// `lds_byte`. D# layout per CDNA5 ISA ch.8. Non-temporal: stream is read-once.
static __device__ __forceinline__ void tdm_load_1d(uint32_t lds_byte,
                                                   const float* gptr,
                                                   uint32_t nelem)
{
    const uint64_t ga = (uint64_t)(uintptr_t)gptr;
    u32x4 g0;
    g0.x = 1u;                                   // count=1, user mode, no gather
    g0.y = lds_byte;                             // lds_addr (bytes)
    g0.z = (uint32_t)ga;                         // global_addr[31:0]
    g0.w = ((uint32_t)(ga >> 32) & 0x01FFFFFFu)  // global_addr[56:32]
         | (2u << 30);                           // type = 2 ("image")
    u32x8 g1;
    g1[0] = 2u << 16;                            // workgroup_mask=0, data_size=4B
    g1[1] = (nelem & 0xFFFFu) << 16;             // tensor_dim0[15:0]  @ 63:48
    g1[2] = (nelem >> 16)                        // tensor_dim0[31:16] @ 79:64
          | (1u << 16);                          // tensor_dim1 = 1
    g1[3] = (nelem & 0xFFFFu) << 16;             // tile_dim0 @ 127:112
    g1[4] = 0u;                                  // tile_dim1/2 = 0 (unused)
    g1[5] = nelem;                               // dim0 stride (single row)
    g1[6] = 0u;
    g1[7] = 0u;
    asm volatile("tensor_load_to_lds %0, %1 th:TH_LOAD_NT"
                 :: "s"(g0), "s"(g1) : "memory");
}

// Shuffle + LDS reduction over 8 wave32s; returns the total to ALL threads.
static __device__ __forceinline__ float block_reduce(float v, float* red8,
                                                     int tid, int wave, int lane)
{
    for (int off = 16; off > 0; off >>= 1)
        v += __shfl_down(v, off, 32);
    __syncthreads();                 // red8 safe to overwrite
    if (lane == 0) red8[wave] = v;
    __syncthreads();
    float s = 0.0f;
    for (int w = 0; w < 8; ++w) s += red8[w];   // broadcast reads
    return s;
}

__global__ __launch_bounds__(256)
void listmle_tail_kernel(const float* __restrict__ scores,
                         const int*   __restrict__ target,
                         const int*   __restrict__ tails,
                         const int*   __restrict__ tail_len,
                         float*       __restrict__ loss,
                         int B, int V, int T)
{
    __shared__ float lbuf[NBUF][CHUNK];
    __shared__ float red[8];
    __shared__ float es_sh[TMAX];

    const int b    = blockIdx.x;
    if (b >= B) return;
    const int tid  = threadIdx.x;
    const int wave = tid >> 5;
    const int lane = tid & 31;

    const float* row = scores + (size_t)b * (size_t)V;
    const int nch = (V + CHUNK - 1) / CHUNK;

    const uint32_t lds_base = (uint32_t)(uintptr_t)(&lbuf[0][0]);

    // Prime: up to 3 chunks in flight (wave 0 owns the TDM).
    if (wave == 0) {
        const int pre = (nch < 3) ? nch : 3;
        for (int k = 0; k < pre; ++k) {
            const int bs = k * CHUNK;
            const int nk = (V - bs < CHUNK) ? (V - bs) : CHUNK;
            tdm_load_1d(lds_base + (uint32_t)(k & (NBUF - 1)) * (CHUNK * 4u),
                        row + bs, (uint32_t)nk);
        }
    }

    float acc = 0.0f;
    for (int c = 0; c < nch; ++c) {
        if (wave == 0) {
            // outstanding = min(nch, c+3) - c; wait until chunk c complete.
            if (c + 3 <= nch)      __builtin_amdgcn_s_wait_tensorcnt(2);
            else if (c + 2 <= nch) __builtin_amdgcn_s_wait_tensorcnt(1);
            else                   __builtin_amdgcn_s_wait_tensorcnt(0);
        }
        __syncthreads();   // chunk c ready; also: all waves done with buf[(c-1)%4]

        // Refill the buffer everyone just vacated (WAR-safe after the barrier).
        if (wave == 0 && c + 3 < nch) {
            const int bs = (c + 3) * CHUNK;
            const int nk = (V - bs < CHUNK) ? (V - bs) : CHUNK;
            tdm_load_1d(lds_base + (uint32_t)((c + 3) & (NBUF - 1)) * (CHUNK * 4u),
                        row + bs, (uint32_t)nk);
        }

        const int n  = (V - c * CHUNK < CHUNK) ? (V - c * CHUNK) : CHUNK;
        const int n4 = n >> 2;
        const float4* buf4 = (const float4*)&lbuf[c & (NBUF - 1)][0];
        for (int i = tid; i < n4; i += 256) {
            const float4 v = buf4[i];
            acc += __expf(v.x) + __expf(v.y) + __expf(v.z) + __expf(v.w);
        }
        if ((n & 3) && tid == 0) {           // generality; V=50000 is 4-aligned
            const float* bs = (const float*)buf4;
            for (int i = n4 * 4; i < n; ++i) acc += __expf(bs[i]);
        }
        // no trailing barrier: next iteration's barrier provides WAR protection
    }

    const float total = block_reduce(acc, red, tid, wave, lane);

    // ---- Tail term, parallel across lanes (L <= T-1 = 49) ----
    int L = tail_len[b];                  // scalar (uniform) load
    if (L > T) L = T;

    float sv = 0.0f, e = 0.0f;
    if (tid < L) {
        const int col = tails[b * T + tid];   // one parallel gather round-trip
        sv = row[col];
        e  = __expf(sv);
        es_sh[tid] = e;
    }
    __syncthreads();

    const float above  = block_reduce((tid < L) ? sv : 0.0f, red, tid, wave, lane);
    const float sum_es = block_reduce((tid < L) ? e  : 0.0f, red, tid, wave, lane);

    const float ts     = row[target[b]];      // uniform load
    const float others = total - __expf(ts) - sum_es;

    // suffix_t = sum_{u>=t} es[u]; reversed-cumsum-of-logs, all lanes parallel
    float lg = 0.0f;
    if (tid < L) {
        float suffix = 0.0f;
        for (int u = tid; u < L; ++u) suffix += es_sh[u];
        lg = __logf(others + suffix);
    }
    const float below = block_reduce(lg, red, tid, wave, lane);

    if (tid == 0) {
        const float tail_term = (L > 0) ? (above - below) : 0.0f;
        loss[b] = -((ts - __logf(total)) + tail_term);
    }
}

extern "C" void kernel_launch(void* const* d_in, const int* in_sizes, int n_in,
                              void* d_out, int out_size, void* d_ws, size_t ws_size,
                              hipStream_t stream) {
    (void)n_in; (void)d_ws; (void)ws_size; (void)out_size;
    const float* scores   = (const float*)d_in[0];
    const int*   target   = (const int*)d_in[1];
    const int*   tails    = (const int*)d_in[2];
    const int*   tail_len = (const int*)d_in[3];
    float*       out      = (float*)d_out;

    const int B = in_sizes[1];              // (B,) target
    const int V = in_sizes[0] / B;          // (B,V) output
    const int T = in_sizes[2] / B;          // (B,T) tails

    listmle_tail_kernel<<<dim3(B), dim3(256), 0, stream>>>(
        scores, target, tails, tail_len, out, B, V, T);
}